// AnatomicalStructureEncoder_41197326303846
// MI455X (gfx1250) — compile-verified
//
#include <hip/hip_runtime.h>
#include <cstdint>

#define NREG   20
#define EMBED  768
#define BATCH  2
#define NPATCH 16384
#define DD 64
#define HH 512
#define WW 512
#define PPW 4   // patches per wave (double-buffered TDM pipeline)

typedef float v2f  __attribute__((ext_vector_type(2)));
typedef float v8f  __attribute__((ext_vector_type(8)));
typedef unsigned int u32x4 __attribute__((ext_vector_type(4)));
typedef int i32x4 __attribute__((ext_vector_type(4)));
typedef int i32x8 __attribute__((ext_vector_type(8)));

__device__ __forceinline__ unsigned rfl(unsigned v) {
  return (unsigned)__builtin_amdgcn_readfirstlane((int)v);
}

struct Geo { int ex, ey, ez; };

// Build a per-patch Tensor-DMA descriptor (D#) and issue TENSOR_LOAD_TO_LDS:
// 3D tile (x=16, y=16, z=4) of 4-byte elements from the (64,512,512) volume,
// starting at the clipped box origin. OOB-beyond-tensor reads return zero.
__device__ __forceinline__ Geo tdm_issue(const int* __restrict__ seg,
                                         const float* __restrict__ pc,
                                         int p, unsigned ldsOff) {
  const int b = p >> 14;  // 16384 patches per batch
  const float cz = pc[3 * (size_t)p + 0] * 64.0f;
  const float cy = pc[3 * (size_t)p + 1] * 512.0f;
  const float cx = pc[3 * (size_t)p + 2] * 512.0f;
  const int z0 = max(0, (int)floorf(cz - 2.0f));
  const int y0 = max(0, (int)floorf(cy - 8.0f));
  const int x0 = max(0, (int)floorf(cx - 8.0f));
  const int z1 = (int)fminf(64.0f,  floorf(cz + 2.0f));
  const int y1 = (int)fminf(512.0f, floorf(cy + 8.0f));
  const int x1 = (int)fminf(512.0f, floorf(cx + 8.0f));
  Geo g; g.ex = max(x1 - x0, 0); g.ey = max(y1 - y0, 0); g.ez = max(z1 - z0, 0);

  const unsigned long long lin =
      (unsigned long long)b * (unsigned long long)DD * HH * WW
    + (unsigned long long)z0 * (HH * WW)
    + (unsigned long long)y0 * WW
    + (unsigned long long)x0;
  const unsigned long long gaddr = (unsigned long long)(uintptr_t)seg + 4ull * lin;

  u32x4 g0;
  g0[0] = rfl(1u);                       // count=1, no gather mode
  g0[1] = rfl(ldsOff);                   // LDS destination (bytes)
  g0[2] = rfl((unsigned)gaddr);          // global_addr[31:0]
  g0[3] = rfl(((unsigned)(gaddr >> 32) & 0x1FFFFFFu) | 0x80000000u); // [56:32] | type=2
  i32x8 g1;
  g1[0] = 0x00020000;  // workgroup_mask=0, data_size=2 (4B)
  g1[1] = 0x02000000;  // tensor_dim0 = 512 (low16 in [31:16])
  g1[2] = 0x02000000;  // tensor_dim0 hi=0, tensor_dim1 = 512 (low16 in [31:16])
  g1[3] = 0x00100000;  // tensor_dim1 hi=0, tile_dim0 = 16
  g1[4] = 0x00040010;  // tile_dim1 = 16, tile_dim2 = 4
  g1[5] = 512;         // tensor_dim0_stride low32 (= W)
  g1[6] = 0;           // dim0_stride hi, dim1_stride low16 (0x40000 & 0xffff = 0)
  g1[7] = 4;           // tensor_dim1_stride bits[47:16] (0x40000 >> 16 = 4 -> 262144)
  i32x4 g2; g2[0] = DD; g2[1] = 0; g2[2] = 0; g2[3] = 0;  // tensor_dim2 = 64
  i32x4 g3; g3[0] = 0;  g3[1] = 0; g3[2] = 0; g3[3] = 0;
#if defined(__clang_major__) && (__clang_major__ >= 23)
  i32x8 g4 = {};
  __builtin_amdgcn_tensor_load_to_lds(g0, g1, g2, g3, g4, 0);
#else
  __builtin_amdgcn_tensor_load_to_lds(g0, g1, g2, g3, 0);
#endif
  return g;
}

__device__ __forceinline__ void process_tile(const int* __restrict__ tile,
                                             float* hist, Geo g, int p,
                                             float* __restrict__ out2, int lane) {
  // Zero 32x32 f32 histogram (rows = raw voxel value 0..31, cols = lane).
  float4* h4 = reinterpret_cast<float4*>(hist);
  const float4 z4 = make_float4(0.f, 0.f, 0.f, 0.f);
#pragma unroll
  for (int j = 0; j < 8; ++j) h4[j * 32 + lane] = z4;

  // Lane L owns linear voxels [32L, 32L+32): z = L>>3, two y-rows, x = i&15.
  const int zz = lane >> 3;
  const int ya = (2 * lane) & 15;
  const bool rv0 = (zz < g.ez) && (ya     < g.ey);
  const bool rv1 = (zz < g.ez) && (ya + 1 < g.ey);
  const int4* t4 = reinterpret_cast<const int4*>(tile);
#pragma unroll
  for (int j = 0; j < 8; ++j) {
    const int4 v = t4[lane * 8 + j];
    const bool rv = (j < 4) ? rv0 : rv1;
    const int xb = (j & 3) * 4;
    const int m0 = (rv && (xb + 0 < g.ex)) ? v.x : 0;  // invalid -> row 0 (uncounted)
    const int m1 = (rv && (xb + 1 < g.ex)) ? v.y : 0;
    const int m2 = (rv && (xb + 2 < g.ex)) ? v.z : 0;
    const int m3 = (rv && (xb + 3 < g.ex)) ? v.w : 0;
    atomicAdd(&hist[m0 * 32 + lane], 1.0f);  // conflict-free: column == lane
    atomicAdd(&hist[m1 * 32 + lane], 1.0f);
    atomicAdd(&hist[m2 * 32 + lane], 1.0f);
    atomicAdd(&hist[m3 * 32 + lane], 1.0f);
  }

  // Cross-lane reduce with v_wmma_f32_16x16x4_f32: D = sum_c A_c * ones.
  // A (16x4 f32): lane m (0-15) holds [M=m, K=c4+0..1], lanes 16-31 hold K=c4+2..3.
  v2f ones; ones[0] = 1.0f; ones[1] = 1.0f;
  const int rowb = lane & 15;
  const int kh = (lane >> 4) << 1;
  v8f acc0 = {}; v8f acc1 = {};
#pragma unroll
  for (int c = 0; c < 8; ++c) {
    v2f a0 = *reinterpret_cast<const v2f*>(&hist[rowb * 32 + c * 4 + kh]);
    acc0 = __builtin_amdgcn_wmma_f32_16x16x4_f32(false, a0, false, ones,
                                                 (short)0, acc0, false, false);
    v2f a1 = *reinterpret_cast<const v2f*>(&hist[(16 + rowb) * 32 + c * 4 + kh]);
    acc1 = __builtin_amdgcn_wmma_f32_16x16x4_f32(false, a1, false, ones,
                                                 (short)0, acc1, false, false);
  }
  // D layout: lanes 0-15 VGPR j = S[j] (set0) / S[16+j] (set1); lanes 16-31: S[8+j].
  float* Sv = hist;  // reuse row 0..23 as the S[] scatter area
  if (lane == 0) {
    h4[0] = make_float4(acc0[0], acc0[1], acc0[2], acc0[3]);   // S[0..3]
    h4[1] = make_float4(acc0[4], acc0[5], acc0[6], acc0[7]);   // S[4..7]
    h4[4] = make_float4(acc1[0], acc1[1], acc1[2], acc1[3]);   // S[16..19]
    h4[5] = make_float4(acc1[4], acc1[5], acc1[6], acc1[7]);   // S[20..23]
  } else if (lane == 16) {
    h4[2] = make_float4(acc0[0], acc0[1], acc0[2], acc0[3]);   // S[8..11]
    h4[3] = make_float4(acc0[4], acc0[5], acc0[6], acc0[7]);   // S[12..15]
  }
  const float cnt = Sv[(lane < NREG) ? (lane + 1) : 21];  // region r <-> value r+1
  const float numel = (float)(g.ex * g.ey * g.ez);
  const float inv = 1.0f / fmaxf(numel, 1.0f);
  const float a = (lane < NREG) ? (cnt * inv + 1e-6f) : 0.0f;
  float t = a;
  t += __shfl_xor(t, 1, 32);
  t += __shfl_xor(t, 2, 32);
  t += __shfl_xor(t, 4, 32);
  t += __shfl_xor(t, 8, 32);
  t += __shfl_xor(t, 16, 32);
  const float o = a / t;
  if (lane < NREG) out2[(size_t)p * NREG + lane] = o;
}

__global__ __launch_bounds__(32)
void patch_hist(const int* __restrict__ seg, const float* __restrict__ pc,
                float* __restrict__ out2) {
  __shared__ int   tile0[1024];
  __shared__ int   tile1[1024];
  __shared__ float hist[1024];
  const int lane = threadIdx.x;
  const int pbase = blockIdx.x * PPW;
  const unsigned off0 = rfl((unsigned)(uintptr_t)tile0);
  const unsigned off1 = rfl((unsigned)(uintptr_t)tile1);

  Geo g[2];
  g[0] = tdm_issue(seg, pc, pbase, off0);        // prime the pipeline
#pragma unroll
  for (int j = 0; j < PPW; ++j) {
    const int buf = j & 1;
    if (j + 1 < PPW) {                            // prefetch next tile via TDM
      g[(j + 1) & 1] = tdm_issue(seg, pc, pbase + j + 1, (buf == 0) ? off1 : off0);
      __builtin_amdgcn_s_wait_tensorcnt(1);       // oldest (current) TDM complete
    } else {
      __builtin_amdgcn_s_wait_tensorcnt(0);
    }
    process_tile(buf ? tile1 : tile0, hist, g[buf], pbase + j, out2, lane);
  }
}

__global__ void proto_bcast(const float* __restrict__ proto, float* __restrict__ out) {
  const int i = blockIdx.x * 256 + threadIdx.x;
  const int n1 = NREG * EMBED;
  if (i < BATCH * n1) out[i] = proto[(i >= n1) ? (i - n1) : i];
}

extern "C" void kernel_launch(void* const* d_in, const int* in_sizes, int n_in,
                              void* d_out, int out_size, void* d_ws, size_t ws_size,
                              hipStream_t stream) {
  const int*   seg   = (const int*)d_in[0];
  const float* pc    = (const float*)d_in[1];
  const float* proto = (const float*)d_in[2];
  float* out = (float*)d_out;

  proto_bcast<<<(BATCH * NREG * EMBED + 255) / 256, 256, 0, stream>>>(proto, out);
  patch_hist<<<(BATCH * NPATCH) / PPW, 32, 0, stream>>>(seg, pc,
                                                        out + BATCH * NREG * EMBED);
  (void)in_sizes; (void)n_in; (void)out_size; (void)d_ws; (void)ws_size;
}